// GraphAttention_82738249990883
// MI455X (gfx1250) — compile-verified
//
#include <hip/hip_runtime.h>
#include <hip/hip_bf16.h>

typedef __attribute__((ext_vector_type(2))) float v2f;
typedef __attribute__((ext_vector_type(8))) float v8f;

#define NODE_DIM 128
#define EDGE_DIM 64
#define NHEAD 8
#define DKv 16

// ---------------------------------------------------------------------------
// Zero-init kernel (denom + agg live in workspace which the harness poisons)
// ---------------------------------------------------------------------------
__global__ __launch_bounds__(256) void k_zero(float* __restrict__ p, int n) {
    int i = blockIdx.x * 256 + threadIdx.x;
    if (i < n) p[i] = 0.0f;
}

// ---------------------------------------------------------------------------
// Fused QKV GEMM: reads the 16-row node stripe ONCE, produces Q,K,V.
// One block = 16-row stripe; 8 waves each own one 16-col tile per output.
// ---------------------------------------------------------------------------
__global__ __launch_bounds__(256) void k_qkv_gemm(
    const float* __restrict__ A,
    const float* __restrict__ wq, const float* __restrict__ bq,
    const float* __restrict__ wk, const float* __restrict__ bk,
    const float* __restrict__ wv, const float* __restrict__ bv,
    float* __restrict__ Qo, float* __restrict__ Ko, float* __restrict__ Vo,
    int M)
{
    __shared__ float sA[16 * NODE_DIM];  // 8 KB
    const int m0 = blockIdx.x * 16;
    const int t  = threadIdx.x;

    for (int i = t; i < 16 * NODE_DIM; i += 256) {
        int r = i >> 7, c = i & 127;
        int gr = m0 + r;
        sA[i] = (gr < M) ? A[(size_t)gr * NODE_DIM + c] : 0.0f;
    }
    __syncthreads();

    const int wave = t >> 5;
    const int lane = t & 31;
    const int hl   = lane >> 4;
    const int l16  = lane & 15;
    const int n0   = wave * 16;

    const float* Ws[3] = {wq, wk, wv};
    const float* Bs[3] = {bq, bk, bv};
    float*       Os[3] = {Qo, Ko, Vo};

    for (int m = 0; m < 3; ++m) {
        const float* W = Ws[m];
        v8f acc = {};
        for (int kk = 0; kk < NODE_DIM; kk += 4) {
            const int ka = kk + hl * 2;
            v2f a, b;
            a.x = sA[l16 * NODE_DIM + ka];
            a.y = sA[l16 * NODE_DIM + ka + 1];
            b.x = W[(size_t)(ka)     * NODE_DIM + n0 + l16];
            b.y = W[(size_t)(ka + 1) * NODE_DIM + n0 + l16];
            acc = __builtin_amdgcn_wmma_f32_16x16x4_f32(false, a, false, b,
                                                        (short)0, acc, false, false);
        }
        const float bias = Bs[m][n0 + l16];
        float* O = Os[m];
#pragma unroll
        for (int i = 0; i < 8; ++i) {
            int row = m0 + hl * 8 + i;
            if (row < M) O[(size_t)row * NODE_DIM + n0 + l16] = acc[i] + bias;
        }
    }
}

// ---------------------------------------------------------------------------
// Fused edge kernel (persistent blocks over 16-edge tiles) with
// double-buffered ASYNC staging of edge_attr tiles into LDS:
//   Phase A: ef = edge_attr @ we + be  (16x128 tile via WMMA into LDS)
//   Phase B: per-head scores, exp, atomic segment-sums of denom and es*V
// ---------------------------------------------------------------------------
__global__ __launch_bounds__(256) void k_edge(
    const float* __restrict__ edge_attr,      // [E,64]
    const long long* __restrict__ eidx,       // [2,E]
    const float* __restrict__ we,             // [64,128]
    const float* __restrict__ be,             // [128]
    const float* __restrict__ Q,
    const float* __restrict__ Kt,
    const float* __restrict__ V,
    float* __restrict__ denom,                // [N,8]
    float* __restrict__ agg,                  // [N,128]
    int E, int N)
{
    __shared__ float sEA[2][16 * EDGE_DIM];   // 2 x 4 KB double buffer
    __shared__ float sEF[16 * NODE_DIM];      // 8 KB

    const int t    = threadIdx.x;
    const int wave = t >> 5;
    const int lane = t & 31;
    const int hl   = lane >> 4;
    const int l16  = lane & 15;
    const int numTiles = (E + 15) >> 4;
    const int stride   = gridDim.x;
    const float inv_sqrt_dk = 0.25f;

    // LDS byte offset = low 32 bits of the generic shared-memory address
    // (ISA 10.2, LDS aperture).
    const unsigned dstBase = (unsigned)(size_t)(&sEA[0][0]) + (unsigned)t * 16u;
    const unsigned bufStride = 16u * EDGE_DIM * 4u;  // 4 KB

    int tile = blockIdx.x;
    if (tile >= numTiles) return;

    // Stage a tile into buffer b. Returns true if issued asynchronously.
    auto stage = [&](int tl, int b) -> bool {
        const int e0 = tl * 16;
        if (e0 + 16 <= E) {
            // Each of 256 threads copies 16 contiguous bytes: tile is 4 KB.
            const float* g = edge_attr + (size_t)e0 * EDGE_DIM + t * 4;
            const unsigned dst = dstBase + (unsigned)b * bufStride;
            asm volatile("global_load_async_to_lds_b128 %0, %1, off"
                         :: "v"(dst), "v"(g) : "memory");
            return true;
        }
        // tail tile: guarded synchronous load
        for (int i = t; i < 16 * EDGE_DIM; i += 256) {
            int r = i >> 6, c = i & 63;
            int ge = e0 + r;
            sEA[b][i] = (ge < E) ? edge_attr[(size_t)ge * EDGE_DIM + c] : 0.0f;
        }
        return false;
    };

    int  buf      = 0;
    bool curAsync = stage(tile, 0);

    for (; tile < numTiles; tile += stride) {
        const int e0 = tile * 16;
        const int nt = tile + stride;
        bool nxtAsync = false;
        if (nt < numTiles) nxtAsync = stage(nt, buf ^ 1);

        // Retire the current buffer's async load (loads complete in order:
        // after this wait only the just-issued next-tile load is outstanding).
        if (curAsync) {
            if (nxtAsync) asm volatile("s_wait_asynccnt 0x1" ::: "memory");
            else          asm volatile("s_wait_asynccnt 0x0" ::: "memory");
        }
        __syncthreads();

        const float* sA = sEA[buf];

        // Phase A: ef tile = sA[16x64] @ we[64x128] + be  -> sEF
        {
            const int n0 = wave * 16;
            v8f acc = {};
            for (int kk = 0; kk < EDGE_DIM; kk += 4) {
                const int ka = kk + hl * 2;
                v2f a, b;
                a.x = sA[l16 * EDGE_DIM + ka];
                a.y = sA[l16 * EDGE_DIM + ka + 1];
                b.x = we[(size_t)(ka)     * NODE_DIM + n0 + l16];
                b.y = we[(size_t)(ka + 1) * NODE_DIM + n0 + l16];
                acc = __builtin_amdgcn_wmma_f32_16x16x4_f32(false, a, false, b,
                                                            (short)0, acc, false, false);
            }
            const float bv = be[n0 + l16];
#pragma unroll
            for (int i = 0; i < 8; ++i)
                sEF[(hl * 8 + i) * NODE_DIM + n0 + l16] = acc[i] + bv;
        }
        __syncthreads();

        // Phase B: 16 threads per edge; thread j owns dk=j across all 8 heads
        {
            const int el = t >> 4;
            const int j  = t & 15;
            const int ge = e0 + el;
            if (ge < E) {
                const int r = (int)eidx[ge];        // destination (row)
                const int c = (int)eidx[E + ge];    // source (col)
                const float* qr = Q  + (size_t)r * NODE_DIM;
                const float* kc = Kt + (size_t)c * NODE_DIM;
                const float* vc = V  + (size_t)c * NODE_DIM;

                float p[NHEAD];
#pragma unroll
                for (int h = 0; h < NHEAD; ++h) {
                    int idx = h * DKv + j;
                    p[h] = qr[idx] * (kc[idx] + sEF[el * NODE_DIM + idx]);
                }
#pragma unroll
                for (int off = 8; off >= 1; off >>= 1) {
#pragma unroll
                    for (int h = 0; h < NHEAD; ++h)
                        p[h] += __shfl_xor(p[h], off, 16);
                }
                float es[NHEAD];
#pragma unroll
                for (int h = 0; h < NHEAD; ++h)
                    es[h] = __expf(p[h] * inv_sqrt_dk);

                if (j < NHEAD)
                    atomicAdd(&denom[(size_t)r * NHEAD + j], es[j]);
#pragma unroll
                for (int h = 0; h < NHEAD; ++h) {
                    int idx = h * DKv + j;
                    atomicAdd(&agg[(size_t)r * NODE_DIM + idx], es[h] * vc[idx]);
                }
            }
        }
        __syncthreads();  // protect LDS reuse next iteration

        curAsync = nxtAsync;
        buf ^= 1;
    }
}

// ---------------------------------------------------------------------------
// Output GEMM with fused normalization:
//   out = (agg / (denom + 1e-8)) @ wo + bo
// ---------------------------------------------------------------------------
__global__ __launch_bounds__(256) void k_out_gemm(
    const float* __restrict__ agg, const float* __restrict__ denom,
    const float* __restrict__ W, const float* __restrict__ bias,
    float* __restrict__ out, int M)
{
    __shared__ float sA[16 * NODE_DIM];
    const int m0 = blockIdx.x * 16;
    const int t  = threadIdx.x;

    for (int i = t; i < 16 * NODE_DIM; i += 256) {
        int r = i >> 7, c = i & 127;
        int gr = m0 + r;
        float v = 0.0f;
        if (gr < M) {
            float d = denom[(size_t)gr * NHEAD + (c >> 4)] + 1e-8f;
            v = agg[(size_t)gr * NODE_DIM + c] / d;
        }
        sA[i] = v;
    }
    __syncthreads();

    const int wave = t >> 5;
    const int lane = t & 31;
    const int hl   = lane >> 4;
    const int l16  = lane & 15;
    const int n0   = wave * 16;

    v8f acc = {};
    for (int kk = 0; kk < NODE_DIM; kk += 4) {
        const int ka = kk + hl * 2;
        v2f a, b;
        a.x = sA[l16 * NODE_DIM + ka];
        a.y = sA[l16 * NODE_DIM + ka + 1];
        b.x = W[(size_t)(ka)     * NODE_DIM + n0 + l16];
        b.y = W[(size_t)(ka + 1) * NODE_DIM + n0 + l16];
        acc = __builtin_amdgcn_wmma_f32_16x16x4_f32(false, a, false, b,
                                                    (short)0, acc, false, false);
    }
    const float bv = bias[n0 + l16];
#pragma unroll
    for (int i = 0; i < 8; ++i) {
        int row = m0 + hl * 8 + i;
        if (row < M) out[(size_t)row * NODE_DIM + n0 + l16] = acc[i] + bv;
    }
}

// ---------------------------------------------------------------------------
extern "C" void kernel_launch(void* const* d_in, const int* in_sizes, int n_in,
                              void* d_out, int out_size, void* d_ws, size_t ws_size,
                              hipStream_t stream) {
    const float*     nodes     = (const float*)d_in[0];
    const long long* eidx      = (const long long*)d_in[1];
    const float*     edge_attr = (const float*)d_in[2];
    const float*     wq = (const float*)d_in[3];
    const float*     bq = (const float*)d_in[4];
    const float*     wk = (const float*)d_in[5];
    const float*     bk = (const float*)d_in[6];
    const float*     wv = (const float*)d_in[7];
    const float*     bv = (const float*)d_in[8];
    const float*     we = (const float*)d_in[9];
    const float*     be = (const float*)d_in[10];
    const float*     wo = (const float*)d_in[11];
    const float*     bo = (const float*)d_in[12];

    const int N = in_sizes[0] / NODE_DIM;
    const int E = in_sizes[1] / 2;

    // Workspace layout (floats): Q | K | V | agg | denom
    float* Q     = (float*)d_ws;
    float* K     = Q + (size_t)N * NODE_DIM;
    float* V     = K + (size_t)N * NODE_DIM;
    float* agg   = V + (size_t)N * NODE_DIM;
    float* denom = agg + (size_t)N * NODE_DIM;

    const int mTiles = (N + 15) / 16;

    // zero agg + denom (contiguous)
    {
        int nz = N * (NODE_DIM + NHEAD);
        k_zero<<<(nz + 255) / 256, 256, 0, stream>>>(agg, nz);
    }

    k_qkv_gemm<<<mTiles, 256, 0, stream>>>(nodes, wq, bq, wk, bk, wv, bv,
                                           Q, K, V, N);

    k_edge<<<4096, 256, 0, stream>>>(edge_attr, eidx, we, be, Q, K, V,
                                     denom, agg, E, N);

    k_out_gemm<<<mTiles, 256, 0, stream>>>(agg, denom, wo, bo, (float*)d_out, N);
}